// RiskGAT_14989435863140
// MI455X (gfx1250) — compile-verified
//
#include <hip/hip_runtime.h>
#include <math.h>

// ---------------------------------------------------------------------------
// GAT (2 layers) for MI455X / gfx1250, wave32.
// Dense GEMMs use V_WMMA_F32_16X16X4_F32; edge softmax uses L2 atomics.
// N (=100000) is an exact multiple of 16, so GEMM tiles are always full:
// no per-lane guards -> clean clause'd stores, EXEC all-1s throughout.
// ---------------------------------------------------------------------------

typedef __attribute__((ext_vector_type(2))) float v2f;
typedef __attribute__((ext_vector_type(8))) float v8f;

#define NEG_SLOPE 0.2f
#define TPB 256

// --- order-preserving float<->uint encoding for atomicMax on floats --------
__device__ __forceinline__ unsigned encf(float f) {
  unsigned u = __float_as_uint(f);
  return (u & 0x80000000u) ? ~u : (u | 0x80000000u);
}
__device__ __forceinline__ float decf(unsigned u) {
  return (u & 0x80000000u) ? __uint_as_float(u ^ 0x80000000u)
                           : __uint_as_float(~u);
}

__device__ __forceinline__ float elu1(float v) {
  return v > 0.f ? v : (__expf(v) - 1.f);
}

// --- fill ------------------------------------------------------------------
__global__ void fill_f32(float* __restrict__ p, float v, long long n) {
  long long i = blockIdx.x * (long long)blockDim.x + threadIdx.x;
  if (i < n) p[i] = v;
}

// --- Layer 1 GEMM: h1[N,64] = x[N,4] @ W1[4,64] ----------------------------
// One wave per (16-row, 16-col) tile; 4 waves/block cover the 64 columns.
__global__ void gemm1_wmma(const float* __restrict__ x,
                           const float* __restrict__ W1,
                           float* __restrict__ h1) {
  const int lane = threadIdx.x & 31;
  const int nt   = threadIdx.x >> 5;   // N-tile 0..3
  const int m0   = blockIdx.x * 16;
  const int n0   = nt * 16;
  const int ln   = lane & 15;
  const int hi   = lane >> 4;
  const int kb   = hi * 2;             // lanes 0-15 -> K0,1 ; lanes 16-31 -> K2,3

  const v2f a = *(const v2f*)(x + (size_t)(m0 + ln) * 4 + kb);
  v2f b;
  b.x = W1[(size_t)kb * 64 + n0 + ln];
  b.y = W1[(size_t)(kb + 1) * 64 + n0 + ln];

  v8f c = {};
  c = __builtin_amdgcn_wmma_f32_16x16x4_f32(false, a, false, b,
                                            (short)0, c, false, false);
  float* out = h1 + (size_t)(m0 + hi * 8) * 64 + n0 + ln;
#pragma unroll
  for (int r = 0; r < 8; ++r) out[(size_t)r * 64] = c[r];
}

// --- Layer 2 GEMM: h2[N,16] = act1[N,64] @ W2[64,16] -----------------------
// One wave per 16-row tile, 16 chained K=4 WMMAs over K=64.
__global__ void gemm2_wmma(const float* __restrict__ A,
                           const float* __restrict__ W2,
                           float* __restrict__ h2, int n) {
  const int lane = threadIdx.x & 31;
  const int w    = threadIdx.x >> 5;
  const int m0   = (blockIdx.x * 4 + w) * 16;
  if (m0 >= n) return;                 // wave-uniform guard (EXEC stays all-1)
  const int ln = lane & 15;
  const int hi = lane >> 4;
  const int kb = hi * 2;

  const float* arow = A + (size_t)(m0 + ln) * 64;

  v8f c = {};
#pragma unroll
  for (int kk = 0; kk < 16; ++kk) {
    const int k0 = kk * 4 + kb;
    const v2f a = *(const v2f*)(arow + k0);
    v2f b;
    b.x = W2[(size_t)k0 * 16 + ln];
    b.y = W2[(size_t)(k0 + 1) * 16 + ln];
    c = __builtin_amdgcn_wmma_f32_16x16x4_f32(false, a, false, b,
                                              (short)0, c, false, false);
  }
  float* out = h2 + (size_t)(m0 + hi * 8) * 16 + ln;
#pragma unroll
  for (int r = 0; r < 8; ++r) out[(size_t)r * 16] = c[r];
}

// --- per-node attention logits: al_s[i,h]=<h_i,a_src[h]>, same for dst -----
template <int H, int D>
__global__ void logits_k(const float* __restrict__ h,
                         const float* __restrict__ a_src,
                         const float* __restrict__ a_dst,
                         float* __restrict__ als, float* __restrict__ ald,
                         int n) {
  int i = blockIdx.x * blockDim.x + threadIdx.x;
  if (i >= n) return;
  const float* hr = h + (size_t)i * H * D;
#pragma unroll
  for (int hh = 0; hh < H; ++hh) {
    float s = 0.f, d = 0.f;
#pragma unroll
    for (int j = 0; j < D; ++j) {
      float v = hr[hh * D + j];
      s += v * a_src[hh * D + j];
      d += v * a_dst[hh * D + j];
    }
    als[(size_t)i * H + hh] = s;
    ald[(size_t)i * H + hh] = d;
  }
}

// --- edge pass 1: segment max over dst (order-encoded atomicMax) -----------
template <int H>
__global__ void edge_max_k(const int* __restrict__ src,
                           const int* __restrict__ dst,
                           const float* __restrict__ als,
                           const float* __restrict__ ald,
                           unsigned* __restrict__ menc, int E, int N) {
  int e = blockIdx.x * blockDim.x + threadIdx.x;
  if (e >= E + N) return;
  int s = e < E ? src[e] : e - E;
  int d = e < E ? dst[e] : e - E;
#pragma unroll
  for (int hh = 0; hh < H; ++hh) {
    float v = als[(size_t)s * H + hh] + ald[(size_t)d * H + hh];
    v = v > 0.f ? v : NEG_SLOPE * v;            // leaky_relu
    atomicMax(&menc[(size_t)d * H + hh], encf(v));
  }
}

// --- edge pass 2: ex = exp(e - m[dst]); denom[dst] += ex -------------------
template <int H>
__global__ void edge_exp_k(const int* __restrict__ src,
                           const int* __restrict__ dst,
                           const float* __restrict__ als,
                           const float* __restrict__ ald,
                           const unsigned* __restrict__ menc,
                           float* __restrict__ ex, float* __restrict__ den,
                           int E, int N) {
  int e = blockIdx.x * blockDim.x + threadIdx.x;
  if (e >= E + N) return;
  int s = e < E ? src[e] : e - E;
  int d = e < E ? dst[e] : e - E;
#pragma unroll
  for (int hh = 0; hh < H; ++hh) {
    float v = als[(size_t)s * H + hh] + ald[(size_t)d * H + hh];
    v = v > 0.f ? v : NEG_SLOPE * v;
    float p = __expf(v - decf(menc[(size_t)d * H + hh]));
    ex[(size_t)e * H + hh] = p;
    atomicAdd(&den[(size_t)d * H + hh], p);
  }
}

// --- edge pass 3: agg[dst] += h[src] * alpha (one thread per edge*head) ----
template <int H, int D>
__global__ void edge_agg_k(const int* __restrict__ src,
                           const int* __restrict__ dst,
                           const float* __restrict__ h,
                           const float* __restrict__ ex,
                           const float* __restrict__ den,
                           float* __restrict__ agg, int E, int N) {
  long long t = blockIdx.x * (long long)blockDim.x + threadIdx.x;
  if (t >= (long long)(E + N) * H) return;
  int e  = (int)(t / H);
  int hh = (int)(t % H);
  int s = e < E ? src[e] : e - E;
  int d = e < E ? dst[e] : e - E;
  float alpha = ex[(size_t)e * H + hh] /
                (den[(size_t)d * H + hh] + 1e-16f);
  const float* hs = h + (size_t)s * H * D + hh * D;
  float* ad = agg + (size_t)d * H * D + hh * D;
#pragma unroll
  for (int j = 0; j < D; ++j) atomicAdd(&ad[j], hs[j] * alpha);
}

// --- bias + ELU (in place) -------------------------------------------------
__global__ void elu_bias_k(float* __restrict__ a, const float* __restrict__ b,
                           long long n, int F) {
  long long i = blockIdx.x * (long long)blockDim.x + threadIdx.x;
  if (i >= n) return;
  a[i] = elu1(a[i] + b[(int)(i % F)]);
}

// --- final: sigmoid(elu(agg2 + b2) @ lin_w + lin_b) ------------------------
__global__ void final_k(const float* __restrict__ agg2,
                        const float* __restrict__ b2,
                        const float* __restrict__ lw,
                        const float* __restrict__ lb,
                        float* __restrict__ out, int n) {
  int i = blockIdx.x * blockDim.x + threadIdx.x;
  if (i >= n) return;
  float z = lb[0];
#pragma unroll
  for (int j = 0; j < 16; ++j)
    z += elu1(agg2[(size_t)i * 16 + j] + b2[j]) * lw[j];
  out[i] = 1.f / (1.f + __expf(-z));
}

static inline int cdivll(long long a, int b) { return (int)((a + b - 1) / b); }

extern "C" void kernel_launch(void* const* d_in, const int* in_sizes, int n_in,
                              void* d_out, int out_size, void* d_ws, size_t ws_size,
                              hipStream_t stream) {
  const float* x   = (const float*)d_in[0];
  const int*   ei  = (const int*)d_in[1];
  const float* W1  = (const float*)d_in[2];
  const float* as1 = (const float*)d_in[3];
  const float* ad1 = (const float*)d_in[4];
  const float* b1  = (const float*)d_in[5];
  const float* W2  = (const float*)d_in[6];
  const float* as2 = (const float*)d_in[7];
  const float* ad2 = (const float*)d_in[8];
  const float* b2  = (const float*)d_in[9];
  const float* lw  = (const float*)d_in[10];
  const float* lb  = (const float*)d_in[11];
  float* out = (float*)d_out;

  const int N  = in_sizes[0] / 4;       // 100000 (multiple of 16)
  const int E  = in_sizes[1] / 2;       // 1600000
  const int EN = E + N;
  const int* src = ei;
  const int* dst = ei + E;

  // workspace layout (floats)
  float*    ws   = (float*)d_ws;
  float*    h1   = ws;                               // N*64
  float*    act1 = h1  + (size_t)N * 64;             // N*64 (agg1 -> elu in place)
  float*    als  = act1 + (size_t)N * 64;            // N*4 (layer2 reuses N*2)
  float*    ald  = als + (size_t)N * 4;              // N*4
  unsigned* menc = (unsigned*)(ald + (size_t)N * 4); // N*4
  float*    den  = (float*)(menc + (size_t)N * 4);   // N*4
  float*    ex   = den + (size_t)N * 4;              // (E+N)*4 (layer2 reuses *2)
  float*    h2   = h1;                               // reuse: N*16
  float*    agg2 = h1 + (size_t)N * 16;              // reuse: N*16

  const int mtiles = N / 16;            // exact

  // ===== Layer 1 =====
  gemm1_wmma<<<mtiles, 128, 0, stream>>>(x, W1, h1);
  logits_k<4, 16><<<cdivll(N, TPB), TPB, 0, stream>>>(h1, as1, ad1, als, ald, N);

  fill_f32<<<cdivll((long long)N * 64, TPB), TPB, 0, stream>>>(act1, 0.f, (long long)N * 64);
  fill_f32<<<cdivll((long long)N * 4, TPB), TPB, 0, stream>>>((float*)menc, 0.f, (long long)N * 4);
  fill_f32<<<cdivll((long long)N * 4, TPB), TPB, 0, stream>>>(den, 0.f, (long long)N * 4);

  edge_max_k<4><<<cdivll(EN, TPB), TPB, 0, stream>>>(src, dst, als, ald, menc, E, N);
  edge_exp_k<4><<<cdivll(EN, TPB), TPB, 0, stream>>>(src, dst, als, ald, menc, ex, den, E, N);
  edge_agg_k<4, 16><<<cdivll((long long)EN * 4, TPB), TPB, 0, stream>>>(src, dst, h1, ex, den, act1, E, N);

  elu_bias_k<<<cdivll((long long)N * 64, TPB), TPB, 0, stream>>>(act1, b1, (long long)N * 64, 64);

  // ===== Layer 2 =====
  gemm2_wmma<<<cdivll(mtiles, 4), 128, 0, stream>>>(act1, W2, h2, N);
  logits_k<2, 8><<<cdivll(N, TPB), TPB, 0, stream>>>(h2, as2, ad2, als, ald, N);

  fill_f32<<<cdivll((long long)N * 16, TPB), TPB, 0, stream>>>(agg2, 0.f, (long long)N * 16);
  fill_f32<<<cdivll((long long)N * 2, TPB), TPB, 0, stream>>>((float*)menc, 0.f, (long long)N * 2);
  fill_f32<<<cdivll((long long)N * 2, TPB), TPB, 0, stream>>>(den, 0.f, (long long)N * 2);

  edge_max_k<2><<<cdivll(EN, TPB), TPB, 0, stream>>>(src, dst, als, ald, menc, E, N);
  edge_exp_k<2><<<cdivll(EN, TPB), TPB, 0, stream>>>(src, dst, als, ald, menc, ex, den, E, N);
  edge_agg_k<2, 8><<<cdivll((long long)EN * 2, TPB), TPB, 0, stream>>>(src, dst, h2, ex, den, agg2, E, N);

  // ===== Head =====
  final_k<<<cdivll(N, TPB), TPB, 0, stream>>>(agg2, b2, lw, lb, out, N);
}